// LineEnergyHead7x7_6691559047411
// MI455X (gfx1250) — compile-verified
//
#include <hip/hip_runtime.h>
#include <math.h>
#include <stdint.h>

#define BB 8
#define CC 64
#define HH 320
#define WW 320
#define QQ 24
#define NN 8

typedef __attribute__((ext_vector_type(2))) float v2f;
typedef __attribute__((ext_vector_type(8))) float v8f;
typedef __attribute__((ext_vector_type(4))) uint32_t u32x4;
typedef __attribute__((ext_vector_type(8))) uint32_t u32x8;

// ---------------------------------------------------------------------------
// Pass 1: Fmap[b,h,w] = sqrt(sum_c X[b,c,h,w]^2)
// diag(V^T V) per 16-pixel group via V_WMMA_F32_16X16X4_F32 chained over
// K = 64 channels in full fp32. Two independent accumulator chains (even/odd
// K-chunks) double XDL pipeline overlap; partial Grams summed at the end.
// Diagonal lives in lanes 0..7 (vgpr r = lane) and lanes 24..31 (r = lane-24).
// ---------------------------------------------------------------------------
__global__ __launch_bounds__(128) void norm_kernel(const float* __restrict__ X,
                                                   float* __restrict__ F) {
  const int lane = threadIdx.x & 31;
  const int wv   = threadIdx.x >> 5;
  const int w0   = blockIdx.x * 64 + wv * 16;
  const int h    = blockIdx.y;
  const int b    = blockIdx.z;
  const size_t HW = (size_t)HH * WW;

#if __has_builtin(__builtin_amdgcn_wmma_f32_16x16x4_f32)
  const int half = lane >> 4;  // 0: K={0,1}, 1: K={2,3}
  const int pix  = lane & 15;
  const float* p = X + ((size_t)b * CC + 2 * half) * HW + (size_t)h * WW + w0 + pix;
  v8f acc0 = {0.f, 0.f, 0.f, 0.f, 0.f, 0.f, 0.f, 0.f};
  v8f acc1 = {0.f, 0.f, 0.f, 0.f, 0.f, 0.f, 0.f, 0.f};
#pragma unroll
  for (int c = 0; c < CC; c += 8) {
    v2f a, bvec;
    a.x = __builtin_nontemporal_load(p + (size_t)c * HW);
    a.y = __builtin_nontemporal_load(p + (size_t)(c + 1) * HW);
    bvec.x = __builtin_nontemporal_load(p + (size_t)(c + 4) * HW);
    bvec.y = __builtin_nontemporal_load(p + (size_t)(c + 5) * HW);
    acc0 = __builtin_amdgcn_wmma_f32_16x16x4_f32(false, a, false, a,
                                                 (short)0, acc0, false, false);
    acc1 = __builtin_amdgcn_wmma_f32_16x16x4_f32(false, bvec, false, bvec,
                                                 (short)0, acc1, false, false);
  }
  v8f acc = acc0 + acc1;
  float d = 0.f;
#pragma unroll
  for (int r = 0; r < 8; ++r)
    d = (lane == r || lane == r + 24) ? acc[r] : d;
  if (lane < 8 || lane >= 24) {
    const int m = (lane < 8) ? lane : lane - 16;
    F[(size_t)b * HW + (size_t)h * WW + w0 + m] = sqrtf(d);
  }
#else
  if (lane < 16) {
    const float* p = X + (size_t)b * CC * HW + (size_t)h * WW + w0 + lane;
    float s = 0.f;
    for (int c = 0; c < CC; ++c) {
      float v = __builtin_nontemporal_load(p + (size_t)c * HW);
      s = fmaf(v, v, s);
    }
    F[(size_t)b * HW + (size_t)h * WW + w0 + lane] = sqrtf(s);
  }
#endif
}

// ---------------------------------------------------------------------------
// Pass 2: 192 bilinear samples/pixel from an LDS tile of (x,x+1) float2 pairs
// (one aligned ds_load_b64 per row-pair), Gaussian radial sum -> E_q, online
// softmax folded into unnormalized sin/cos moments (normalization cancels in
// atan2), circular mean, mod 2*pi.
//
// Tile staging: interior tiles use the Tensor Data Mover -- one
// tensor_load_to_lds (2D D#: 16 rows x 44 dwords, tensor stride 320 dwords)
// issued by wave 0, waited with s_wait_tensorcnt, then an LDS pass builds the
// pair array. Edge tiles use a masked global loader for exact zero padding.
// ---------------------------------------------------------------------------
#define TX 32
#define TY 8
#define PW 40
#define PH 16
#define RAWW 44

__global__ __launch_bounds__(256) void orient_kernel(const float* __restrict__ F,
                                                     float* __restrict__ out) {
  __shared__ alignas(16) float raw[PH][RAWW];  // TDM staging (interior tiles)
  __shared__ v2f sp[PH * PW];                  // (x, x+1) pair tile

  const int b   = blockIdx.z;
  const int i0  = blockIdx.y * TY;
  const int w0  = blockIdx.x * TX;
  const int tid = threadIdx.y * TX + threadIdx.x;
  const float* Fb = F + (size_t)b * HH * WW;

  const bool interior = (w0 >= 4) && (w0 <= WW - PW) &&
                        (i0 >= 4) && (i0 <= HH - PH + 4);

  if (interior) {
    // ---- TDM: DMA the raw 16 x 44 tile (rows [i0-4, i0+11], cols [w0-4, w0+39])
    if (threadIdx.y == 0) {
      const uint32_t lds_base =
          (uint32_t)(uintptr_t)(__attribute__((address_space(3))) float*)&raw[0][0];
      const uint64_t gaddr =
          (uint64_t)(uintptr_t)(Fb + (size_t)(i0 - 4) * WW + (w0 - 4));
      const uint32_t td0 = (uint32_t)(WW - (w0 - 4));  // remaining width  (elems)
      const uint32_t td1 = (uint32_t)(HH - (i0 - 4));  // remaining height (elems)

      u32x4 g0;
      g0.x = 1u;                                   // count = 1 (valid descriptor)
      g0.y = lds_base;                             // lds_addr (bytes)
      g0.z = (uint32_t)gaddr;                      // global_addr[31:0]
      g0.w = ((uint32_t)(gaddr >> 32) & 0x01FFFFFFu) | 0x80000000u;  // hi + type=2

      u32x8 g1;
      g1.s0 = 0x00020000u;                         // data_size = 2 (4 bytes)
      g1.s1 = (td0 & 0xFFFFu) << 16;               // tensor_dim0[15:0]
      g1.s2 = (td0 >> 16) | ((td1 & 0xFFFFu) << 16);
      g1.s3 = (td1 >> 16) | ((uint32_t)RAWW << 16);  // tile_dim0 = 44
      g1.s4 = (uint32_t)PH;                        // tile_dim1 = 16, tile_dim2 = 0
      g1.s5 = (uint32_t)WW;                        // tensor_dim0_stride = 320
      g1.s6 = 0u;
      g1.s7 = 0u;

      asm volatile("tensor_load_to_lds %0, %1" ::"s"(g0), "s"(g1) : "memory");
#if __has_builtin(__builtin_amdgcn_s_wait_tensorcnt)
      __builtin_amdgcn_s_wait_tensorcnt(0);
#else
      asm volatile("s_wait_tensorcnt 0x0" ::: "memory");
#endif
    }
    __syncthreads();
    // ---- build (x, x+1) pairs from the raw tile
    for (int s = tid; s < PH * PW; s += TX * TY) {
      const int ry = s / PW;
      const int rx = s - ry * PW;
      v2f pr;
      pr.x = raw[ry][rx];
      pr.y = raw[ry][rx + 1];
      sp[s] = pr;
    }
  } else {
    // ---- edge tiles: masked global loads, zeros outside the image
    for (int s = tid; s < PH * PW; s += TX * TY) {
      const int ry = s / PW;
      const int rx = s - ry * PW;
      const int gy = i0 - 4 + ry;
      const int gx = w0 - 4 + rx;
      float v0 = 0.f, v1 = 0.f;
      if ((unsigned)gy < (unsigned)HH) {
        if ((unsigned)gx < (unsigned)WW)       v0 = Fb[(size_t)gy * WW + gx];
        if ((unsigned)(gx + 1) < (unsigned)WW) v1 = Fb[(size_t)gy * WW + gx + 1];
      }
      v2f pr; pr.x = v0; pr.y = v1;
      sp[s] = pr;
    }
  }
  __syncthreads();

  const int j = w0 + threadIdx.x;
  const int i = i0 + threadIdx.y;

  // Gaussian radial weights w_n, normalized
  float wn[NN];
  {
    float sum = 0.f;
#pragma unroll
    for (int n = 0; n < NN; ++n) {
      const float t = 0.375f * (float)(n + 1);
      wn[n] = __expf(-0.5f * t * t);
      sum += wn[n];
    }
    const float inv = 1.f / sum;
#pragma unroll
    for (int n = 0; n < NN; ++n) wn[n] *= inv;
  }

  const float sxf = (float)WW / (float)(WW - 1);
  const float syf = (float)HH / (float)(HH - 1);
  // tile-local coords: lx = (j - dx)*sx - 0.5 - (w0 - 4)
  const float axc = (float)j * sxf - (float)w0 + 3.5f;
  const float ayc = (float)i * syf - (float)i0 + 3.5f;

  const float twoPi = 6.28318530717958647692f;
  float m = -3.0e38f, Sy = 0.f, Cx = 0.f;

#pragma unroll 1
  for (int q = 0; q < QQ; ++q) {
    const float th = twoPi * (float)q * (1.0f / (float)QQ);
    const float cs = __cosf(th);
    const float sn = __sinf(th);
    float Eq = 0.f;
#pragma unroll
    for (int n = 0; n < NN; ++n) {
      const float t  = 0.375f * (float)(n + 1);
      const float lx = fmaf(-t * cs, sxf, axc);
      const float ly = fmaf(-t * sn, syf, ayc);
      const float fx0 = floorf(lx);
      const float fy0 = floorf(ly);
      const int   x0  = (int)fx0;
      const int   y0  = (int)fy0;
      const float fx  = lx - fx0;
      const float fy  = ly - fy0;
      const int idx = y0 * PW + x0;
      const v2f tp  = sp[idx];        // F(y0,x0..x0+1): one ds_load_b64
      const v2f bt  = sp[idx + PW];   // F(y1,x0..x0+1): one ds_load_b64
      const float top = fmaf(fx, tp.y - tp.x, tp.x);
      const float bot = fmaf(fx, bt.y - bt.x, bt.x);
      Eq = fmaf(wn[n], fmaf(fy, bot - top, top), Eq);
    }
    const float nm = fmaxf(m, Eq);
    const float r  = __expf(m - nm);
    const float e  = __expf(Eq - nm);
    Sy = fmaf(Sy, r, e * sn);
    Cx = fmaf(Cx, r, e * cs);
    m = nm;
  }

  float res = atan2f(Sy, Cx) + twoPi;
  if (res >= twoPi) res -= twoPi;
  out[(size_t)b * HH * WW + (size_t)i * WW + j] = res;
}

// ---------------------------------------------------------------------------
extern "C" void kernel_launch(void* const* d_in, const int* in_sizes, int n_in,
                              void* d_out, int out_size, void* d_ws, size_t ws_size,
                              hipStream_t stream) {
  (void)in_sizes; (void)n_in; (void)out_size; (void)ws_size;
  const float* X    = (const float*)d_in[0];
  float*       out  = (float*)d_out;
  float*       Fmap = (float*)d_ws;  // B*H*W floats = 3.28 MB scratch

  dim3 g1(WW / 64, HH, BB);          // 5 x 320 x 8 blocks, 4 waves x 16 px
  norm_kernel<<<g1, 128, 0, stream>>>(X, Fmap);

  dim3 g2(WW / TX, HH / TY, BB);     // 10 x 40 x 8 tiles
  orient_kernel<<<g2, dim3(TX, TY), 0, stream>>>(Fmap, out);
}